// DetectionLoss_71597104824434
// MI455X (gfx1250) — compile-verified
//
#include <hip/hip_runtime.h>
#include <math.h>

typedef float v2f __attribute__((ext_vector_type(2)));
typedef float v4f __attribute__((ext_vector_type(4)));
typedef float v8f __attribute__((ext_vector_type(8)));

#define B_   8
#define A_   9
#define C_   4
#define H_   256
#define W_   256
#define N_   16
#define HW_  (H_ * W_)

#define BLK            256
#define PIX_PER_THREAD 4
#define PIX_PER_BLOCK  (BLK * PIX_PER_THREAD)   // 1024 pixels (4 rows)
#define BLOCKS_PER_B   (HW_ / PIX_PER_BLOCK)    // 64
#define NBLOCKS        (B_ * BLOCKS_PER_B)      // 512

// ---------------------------------------------------------------------------
// Block-wide sum of one float per thread using V_WMMA_F32_16X16X4_F32.
// A = all-ones 16x4, so D[m][n] = sum_k B[k][n] + C[m][n]. Chaining 4 WMMAs
// over 64 values each sums all 256 values into 16 column sums (rows of D all
// identical). Sum of acc[0] over all 32 lanes is exactly 2x the total.
// Layout-agnostic: any bijective lane<->(k,n) mapping yields the same sum.
// Returns the total in thread 0 (threads 1-31: partial values; threads 32+: 0).
// ---------------------------------------------------------------------------
__device__ __forceinline__ float block_sum_wmma(float v, float* red, int tid) {
  red[tid] = v;
  __syncthreads();
  float total = 0.0f;
  if (tid < 32) {                 // wave 0 only; EXEC all-ones inside
    v2f ones; ones[0] = 1.0f; ones[1] = 1.0f;
    v8f acc = {};
#pragma unroll
    for (int c = 0; c < 4; ++c) {
      v2f bv;
      bv[0] = red[c * 64 + 2 * tid + 0];
      bv[1] = red[c * 64 + 2 * tid + 1];
      acc = __builtin_amdgcn_wmma_f32_16x16x4_f32(
          /*neg_a=*/false, ones, /*neg_b=*/false, bv,
          /*c_mod=*/(short)0, acc, /*reuse_a=*/false, /*reuse_b=*/false);
    }
    float s = acc[0];
#pragma unroll
    for (int off = 16; off >= 1; off >>= 1)
      s += __shfl_down(s, off, 32);
    total = s * 0.5f;             // 32-lane sum counts every column twice
  }
  __syncthreads();
  return total;
}

// ---------------------------------------------------------------------------
// Main streaming kernel: one block = 1024 pixels of one batch image.
// partials[block] = {focal_sum, box_sum, valid_pixel_count}
// ---------------------------------------------------------------------------
__global__ __launch_bounds__(BLK) void det_loss_main(
    const float* __restrict__ cls,     // (B,A,C,H,W)
    const float* __restrict__ bbox,    // (B,A,4,H,W)
    const float* __restrict__ boxes,   // (B,N,4)
    const int*   __restrict__ labels,  // (B,N)
    const float* __restrict__ alpha,   // (C)
    float* __restrict__ partials)      // (NBLOCKS, 3)
{
  __shared__ float sbox[N_][4];
  __shared__ int   sclip[N_][4];
  __shared__ int   slab[N_];
  __shared__ float red[BLK];

  const int tid = threadIdx.x;
  const int b   = blockIdx.x / BLOCKS_PER_B;
  const int blk = blockIdx.x % BLOCKS_PER_B;

  if (tid < N_) {
    float b0 = boxes[(b * N_ + tid) * 4 + 0];
    float b1 = boxes[(b * N_ + tid) * 4 + 1];
    float b2 = boxes[(b * N_ + tid) * 4 + 2];
    float b3 = boxes[(b * N_ + tid) * 4 + 3];
    sbox[tid][0] = b0; sbox[tid][1] = b1; sbox[tid][2] = b2; sbox[tid][3] = b3;
    sclip[tid][0] = min(max((int)b0, 0), W_ - 1);
    sclip[tid][1] = min(max((int)b1, 0), H_ - 1);
    sclip[tid][2] = min(max((int)b2, 0), W_ - 1);
    sclip[tid][3] = min(max((int)b3, 0), H_ - 1);
    slab[tid] = labels[b * N_ + tid];
  }
  __syncthreads();

  const float al0 = alpha[0], al1 = alpha[1], al2 = alpha[2], al3 = alpha[3];

  // This thread's 4 consecutive pixels: row h, columns w0..w0+3.
  const int h  = blk * 4 + (tid >> 6);
  const int w0 = (tid & 63) * 4;
  const int rowoff = h * W_ + w0;        // element offset within one plane

  // ---- rasterize: last covering box wins ----
  int lastn[4] = {-1, -1, -1, -1};
#pragma unroll
  for (int n = 0; n < N_; ++n) {
    const int x1 = sclip[n][0], y1 = sclip[n][1];
    const int x2 = sclip[n][2], y2 = sclip[n][3];
    if (h >= y1 && h <= y2) {
#pragma unroll
      for (int p = 0; p < 4; ++p) {
        const int x = w0 + p;
        if (x >= x1 && x <= x2) lastn[p] = n;
      }
    }
  }
  int   tlab[4];
  bool  valid[4];
  float tbx[4][4];
#pragma unroll
  for (int p = 0; p < 4; ++p) {
    valid[p] = (lastn[p] >= 0);
    const int idx = valid[p] ? lastn[p] : 0;
    tlab[p] = valid[p] ? slab[idx] : 0;
#pragma unroll
    for (int j = 0; j < 4; ++j) tbx[p][j] = sbox[idx][j];
  }

  // ---- stream anchors: focal + SmoothL1 ----
  float facc = 0.0f, bacc = 0.0f;
  const v4f* __restrict__ cls4  = (const v4f*)cls;
  const v4f* __restrict__ bbox4 = (const v4f*)bbox;

  for (int a = 0; a < A_; ++a) {
    float s[4][4];   // [class][pixel]
#pragma unroll
    for (int c = 0; c < C_; ++c) {
      const int e = ((b * A_ + a) * C_ + c) * HW_ + rowoff;
      v4f v = __builtin_nontemporal_load(cls4 + (e >> 2));
      s[c][0] = v[0]; s[c][1] = v[1]; s[c][2] = v[2]; s[c][3] = v[3];
    }
    float pbv[4][4]; // [coord][pixel]
#pragma unroll
    for (int j = 0; j < 4; ++j) {
      const int e = ((b * A_ + a) * 4 + j) * HW_ + rowoff;
      v4f v = __builtin_nontemporal_load(bbox4 + (e >> 2));
      pbv[j][0] = v[0]; pbv[j][1] = v[1]; pbv[j][2] = v[2]; pbv[j][3] = v[3];
    }
#pragma unroll
    for (int p = 0; p < 4; ++p) {
      const float m = fmaxf(fmaxf(s[0][p], s[1][p]), fmaxf(s[2][p], s[3][p]));
      const float e = expf(s[0][p] - m) + expf(s[1][p] - m) +
                      expf(s[2][p] - m) + expf(s[3][p] - m);
      const float lse = m + logf(e);
      const int   t   = tlab[p];
      const float st  = (t == 0) ? s[0][p] : (t == 1) ? s[1][p]
                       : (t == 2) ? s[2][p] : s[3][p];
      const float at  = (t == 0) ? al0 : (t == 1) ? al1
                       : (t == 2) ? al2 : al3;
      const float ce  = lse - st;
      const float pt  = expf(-ce);
      const float om  = 1.0f - pt;
      facc += om * om * at * ce;

      if (valid[p]) {
#pragma unroll
        for (int j = 0; j < 4; ++j) {
          const float d = fabsf(pbv[j][p] - tbx[p][j]);
          bacc += (d < 1.0f) ? 0.5f * d * d : (d - 0.5f);
        }
      }
    }
  }
  const float vc = (valid[0] ? 1.0f : 0.0f) + (valid[1] ? 1.0f : 0.0f) +
                   (valid[2] ? 1.0f : 0.0f) + (valid[3] ? 1.0f : 0.0f);

  // ---- per-block reduction via WMMA ----
  const float fsum = block_sum_wmma(facc, red, tid);
  const float bsum = block_sum_wmma(bacc, red, tid);
  const float csum = block_sum_wmma(vc,   red, tid);
  if (tid == 0) {
    partials[blockIdx.x * 3 + 0] = fsum;
    partials[blockIdx.x * 3 + 1] = bsum;
    partials[blockIdx.x * 3 + 2] = csum;
  }
}

// ---------------------------------------------------------------------------
// Deterministic finalize (tiny, f64 accumulation).
// ---------------------------------------------------------------------------
__global__ void det_loss_final(const float* __restrict__ partials,
                               float* __restrict__ out) {
  if (threadIdx.x == 0 && blockIdx.x == 0) {
    double cls_acc = 0.0, box_acc = 0.0;
    for (int b = 0; b < B_; ++b) {
      double f = 0.0, bx = 0.0, c = 0.0;
      for (int k = 0; k < BLOCKS_PER_B; ++k) {
        const float* p = partials + (b * BLOCKS_PER_B + k) * 3;
        f  += (double)p[0];
        bx += (double)p[1];
        c  += (double)p[2];
      }
      const double cnt = c * (double)(A_ * 4);
      cls_acc += f / (double)(A_ * HW_);
      box_acc += (cnt > 0.0) ? (bx / cnt) : 0.0;
    }
    const float cl = (float)(cls_acc / (double)B_);
    const float bl = (float)(box_acc / (double)B_);
    out[0] = cl + bl;
    out[1] = cl;
    out[2] = bl;
  }
}

extern "C" void kernel_launch(void* const* d_in, const int* in_sizes, int n_in,
                              void* d_out, int out_size, void* d_ws, size_t ws_size,
                              hipStream_t stream) {
  const float* cls    = (const float*)d_in[0];
  const float* bbox   = (const float*)d_in[1];
  const float* boxes  = (const float*)d_in[2];
  const int*   labels = (const int*)d_in[3];
  const float* alpha  = (const float*)d_in[4];
  float* out      = (float*)d_out;
  float* partials = (float*)d_ws;   // NBLOCKS*3 floats = 6 KB

  det_loss_main<<<dim3(NBLOCKS), dim3(BLK), 0, stream>>>(
      cls, bbox, boxes, labels, alpha, partials);
  det_loss_final<<<dim3(1), dim3(32), 0, stream>>>(partials, out);
}